// GINE_9826885173932
// MI455X (gfx1250) — compile-verified
//
#include <hip/hip_runtime.h>

#define HIDDEN   64
#define EDGE_F   4
#define N_LAYERS 4
#define BN_EPS   1e-5f
#define A_STRIDE 68   // padded LDS row stride (floats) to avoid bank conflicts

typedef __attribute__((ext_vector_type(2))) float v2f;
typedef __attribute__((ext_vector_type(8))) float v8f;

// ---------------------------------------------------------------- zero scratch
__global__ void GINE_zero_kernel(float* __restrict__ p, int n) {
    int i = blockIdx.x * blockDim.x + threadIdx.x;
    int stride = gridDim.x * blockDim.x;
    for (; i < n; i += stride) p[i] = 0.0f;
}

// ------------------------------------------------- edge gather / scatter phase
// One wave per edge (grid-stride over edges). Lane owns 2 columns (c0, c0+1).
// e = edge_feat @ We + be is recomputed on the fly (We/be live in registers).
__global__ void GINE_edge_kernel(const float* __restrict__ h,
                                 const float* __restrict__ edge_feat,
                                 const int*   __restrict__ src,
                                 const int*   __restrict__ dst,
                                 const float* __restrict__ We,
                                 const float* __restrict__ be,
                                 float* __restrict__ agg,
                                 int n_edges) {
    const int lane = threadIdx.x & 31;
    const int wid  = threadIdx.x >> 5;
    const int wavesPerBlock = blockDim.x >> 5;
    const int gwave  = blockIdx.x * wavesPerBlock + wid;
    const int nwaves = gridDim.x * wavesPerBlock;
    const int c0 = lane * 2;

    const float2 w0 = *(const float2*)(We + 0 * HIDDEN + c0);
    const float2 w1 = *(const float2*)(We + 1 * HIDDEN + c0);
    const float2 w2 = *(const float2*)(We + 2 * HIDDEN + c0);
    const float2 w3 = *(const float2*)(We + 3 * HIDDEN + c0);
    const float2 bb = *(const float2*)(be + c0);

    for (int e = gwave; e < n_edges; e += nwaves) {
        const int s = src[e];
        const int d = dst[e];
        const float4 ef = *(const float4*)(edge_feat + (size_t)e * EDGE_F);
        const float2 hv = *(const float2*)(h + (size_t)s * HIDDEN + c0);
        float ex = bb.x + ef.x * w0.x + ef.y * w1.x + ef.z * w2.x + ef.w * w3.x;
        float ey = bb.y + ef.x * w0.y + ef.y * w1.y + ef.z * w2.y + ef.w * w3.y;
        float mx = fmaxf(hv.x + ex, 0.0f);
        float my = fmaxf(hv.y + ey, 0.0f);
        float* ap = agg + (size_t)d * HIDDEN + c0;
        atomicAdd(ap,     mx);
        atomicAdd(ap + 1, my);
    }
}

// ----------------------------------------------------- fused MLP (WMMA) phase
// Block = 128 threads (4 waves) handles 16 rows. Wave w computes output
// columns [16w, 16w+16). z = h+agg staged in LDS; y1 = relu(z@W1+b1) staged
// in LDS; y = y1@W2+b2 written to global + per-column sum/sumsq for BN.
__global__ __launch_bounds__(128)
void GINE_mlp_kernel(const float* __restrict__ h,
                     const float* __restrict__ agg,
                     const float* __restrict__ W1,
                     const float* __restrict__ b1,
                     const float* __restrict__ W2,
                     const float* __restrict__ b2,
                     float* __restrict__ y,
                     float* __restrict__ colsum,
                     float* __restrict__ colsumsq,
                     int n_nodes) {
    __shared__ float Alds[16 * A_STRIDE];
    __shared__ float Ylds[16 * A_STRIDE];
    __shared__ float W1lds[HIDDEN * HIDDEN];
    __shared__ float W2lds[HIDDEN * HIDDEN];

    const int t = threadIdx.x;
    const int nbase = blockIdx.x * 16;

    // stage A tile: z = h + agg (16 x 64), coalesced
    for (int i = t; i < 16 * HIDDEN; i += 128) {
        int r = i >> 6, c = i & 63;
        int row = nbase + r;
        if (row >= n_nodes) row = n_nodes - 1;   // clamp (N % 16 == 0 anyway)
        size_t g = (size_t)row * HIDDEN + c;
        Alds[r * A_STRIDE + c] = h[g] + agg[g];
    }
    // stage weights (64 x 64 each)
    for (int i = t; i < HIDDEN * HIDDEN; i += 128) {
        W1lds[i] = W1[i];
        W2lds[i] = W2[i];
    }
    __syncthreads();

    const int wave = t >> 5;
    const int lane = t & 31;
    const int m    = lane & 15;
    const int half = lane >> 4;
    const int n0   = wave * 16;
    const int col  = n0 + m;

    // GEMM1: (h+agg) @ W1  -> relu -> Ylds
    v8f acc = {};
    for (int k0 = 0; k0 < HIDDEN; k0 += 4) {
        v2f a = *(const v2f*)&Alds[m * A_STRIDE + k0 + 2 * half];
        v2f bf;
        bf.x = W1lds[(k0 + 2 * half)     * HIDDEN + col];
        bf.y = W1lds[(k0 + 2 * half + 1) * HIDDEN + col];
        acc = __builtin_amdgcn_wmma_f32_16x16x4_f32(false, a, false, bf,
                                                    (short)0, acc, false, false);
    }
    const float bias1 = b1[col];
#pragma unroll
    for (int i = 0; i < 8; ++i) {
        float v = fmaxf(acc[i] + bias1, 0.0f);     // row = i + 8*half
        Ylds[(i + 8 * half) * A_STRIDE + col] = v;
    }
    __syncthreads();

    // GEMM2: Y @ W2 + b2, store y, accumulate BN column statistics
    v8f acc2 = {};
    for (int k0 = 0; k0 < HIDDEN; k0 += 4) {
        v2f a = *(const v2f*)&Ylds[m * A_STRIDE + k0 + 2 * half];
        v2f bf;
        bf.x = W2lds[(k0 + 2 * half)     * HIDDEN + col];
        bf.y = W2lds[(k0 + 2 * half + 1) * HIDDEN + col];
        acc2 = __builtin_amdgcn_wmma_f32_16x16x4_f32(false, a, false, bf,
                                                     (short)0, acc2, false, false);
    }
    const float bias2 = b2[col];
    float s = 0.0f, ss = 0.0f;
#pragma unroll
    for (int i = 0; i < 8; ++i) {
        float v = acc2[i] + bias2;
        int row = nbase + i + 8 * half;
        if (row < n_nodes) y[(size_t)row * HIDDEN + col] = v;
        s  += v;
        ss += v * v;
    }
    atomicAdd(&colsum[col],   s);
    atomicAdd(&colsumsq[col], ss);
}

// ------------------------------------------ BatchNorm + relu + residual phase
__global__ void GINE_bn_kernel(const float* __restrict__ y,
                               const float* __restrict__ h_in,
                               const float* __restrict__ colsum,
                               const float* __restrict__ colsumsq,
                               const float* __restrict__ gamma,
                               const float* __restrict__ beta,
                               float* __restrict__ h_out,
                               int n_nodes) {
    const int total = n_nodes * HIDDEN;
    const float invN = 1.0f / (float)n_nodes;
    int i = blockIdx.x * blockDim.x + threadIdx.x;
    int stride = gridDim.x * blockDim.x;
    for (; i < total; i += stride) {
        int c = i & (HIDDEN - 1);
        float mu  = colsum[c] * invN;
        float var = colsumsq[c] * invN - mu * mu;
        float inv = rsqrtf(var + BN_EPS);
        float z = gamma[c] * (y[i] - mu) * inv + beta[c];
        h_out[i] = fmaxf(z, 0.0f) + h_in[i];
    }
}

// ---------------------------------------------------------------------- launch
extern "C" void kernel_launch(void* const* d_in, const int* in_sizes, int n_in,
                              void* d_out, int out_size, void* d_ws, size_t ws_size,
                              hipStream_t stream) {
    const float* node_feat = (const float*)d_in[0];
    const float* edge_feat = (const float*)d_in[1];
    const int*   src       = (const int*)d_in[2];
    const int*   dst       = (const int*)d_in[3];
    const float* We        = (const float*)d_in[4];
    const float* be        = (const float*)d_in[5];
    const float* W1        = (const float*)d_in[6];
    const float* b1        = (const float*)d_in[7];
    const float* W2        = (const float*)d_in[8];
    const float* b2        = (const float*)d_in[9];
    const float* gamma     = (const float*)d_in[10];
    const float* beta      = (const float*)d_in[11];

    const int n_nodes = in_sizes[0] / HIDDEN;
    const int n_edges = in_sizes[2];

    // workspace layout (floats): [agg N*64][colsum 64][colsumsq 64][y N*64][hbuf N*64]
    float* ws       = (float*)d_ws;
    float* agg      = ws;
    float* colsum   = agg + (size_t)n_nodes * HIDDEN;
    float* colsumsq = colsum + HIDDEN;
    float* ybuf     = colsumsq + HIDDEN;
    float* hbuf     = ybuf + (size_t)n_nodes * HIDDEN;
    float* outp     = (float*)d_out;

    const int zn = n_nodes * HIDDEN + 2 * HIDDEN;   // agg + both stat arrays
    int zgrid = (zn + 255) / 256;
    if (zgrid > 4096) zgrid = 4096;

    const int mlp_blocks = (n_nodes + 15) / 16;

    for (int l = 0; l < N_LAYERS; ++l) {
        const float* h_in;
        float* h_out;
        switch (l) {                       // ping-pong: ws-buffer <-> d_out
            case 0:  h_in = node_feat; h_out = hbuf; break;
            case 1:  h_in = hbuf;      h_out = outp; break;
            case 2:  h_in = outp;      h_out = hbuf; break;
            default: h_in = hbuf;      h_out = outp; break;
        }

        GINE_zero_kernel<<<zgrid, 256, 0, stream>>>(agg, zn);
        GINE_edge_kernel<<<4096, 256, 0, stream>>>(h_in, edge_feat, src, dst,
                                                   We, be, agg, n_edges);
        GINE_mlp_kernel<<<mlp_blocks, 128, 0, stream>>>(
            h_in, agg,
            W1 + (size_t)l * HIDDEN * HIDDEN, b1 + (size_t)l * HIDDEN,
            W2 + (size_t)l * HIDDEN * HIDDEN, b2 + (size_t)l * HIDDEN,
            ybuf, colsum, colsumsq, n_nodes);
        GINE_bn_kernel<<<4096, 256, 0, stream>>>(ybuf, h_in, colsum, colsumsq,
                                                 gamma + (size_t)l * HIDDEN,
                                                 beta + (size_t)l * HIDDEN,
                                                 h_out, n_nodes);
    }
}